// AMRComEnc_63608465653858
// MI455X (gfx1250) — compile-verified
//
#include <hip/hip_runtime.h>
#include <math.h>

#define N_NODES   100000
#define N_EDGES   1600000
#define DIM       256
#define NUM_SLOTS 800   // 16 * 50

typedef float v2f __attribute__((ext_vector_type(2)));
typedef float v8f __attribute__((ext_vector_type(8)));

// ---------------------------------------------------------------------------
// Repack W (256x256 row-major) into K-pair-packed form:
//   Wp[k2 * 256 + n] = ( W[2*k2][n], W[2*k2+1][n] ),  k2 = 0..127
// so each WMMA B fragment is one 8-byte load.
// ---------------------------------------------------------------------------
__global__ __launch_bounds__(256) void pack_W(const float* __restrict__ W,
                                              v2f* __restrict__ Wp)
{
  int idx = blockIdx.x * blockDim.x + threadIdx.x;   // 128*256 = 32768
  if (idx < (DIM / 2) * DIM) {
    int k2 = idx >> 8;
    int n  = idx & (DIM - 1);
    v2f p = { W[(size_t)(2 * k2) * DIM + n], W[(size_t)(2 * k2 + 1) * DIM + n] };
    Wp[idx] = p;
  }
}

// ---------------------------------------------------------------------------
// GEMM + bias: z[M x 256] = x[M x 256] @ W[256 x 256] + bias
// One wave computes a 16x64 output tile via V_WMMA_F32_16X16X4_F32 (fp32 exact).
// A fragment (16x4): lane l<16 -> A[l][k0+0..1], lane l>=16 -> A[l-16][k0+2..3]
// B fragment (4x16): lane n (&15), vgpr v -> B[v + 2*(lane>>4)][n]
// C/D (16x16):       vgpr r, lane -> D[r + 8*(lane>>4)][lane&15]
// ---------------------------------------------------------------------------
__global__ __launch_bounds__(256) void gemm_bias_wmma(
    const float* __restrict__ x, const v2f* __restrict__ Wp,
    const float* __restrict__ bias, float* __restrict__ z, int M)
{
  const int lane  = threadIdx.x & 31;
  const int gw    = blockIdx.x * 8 + (threadIdx.x >> 5);
  const int mtile = gw >> 2;     // 16 rows per tile
  const int ngrp  = gw & 3;      // 64 cols per group
  if (mtile * 16 >= M) return;   // wave-uniform: EXEC stays all-ones for WMMA

  const int   hi    = lane >> 4;                 // 0 or 1
  const int   row   = mtile * 16 + (lane & 15);
  const int   ncol0 = ngrp * 64 + (lane & 15);
  const float* __restrict__ xrow = x + (size_t)row * DIM;

  v8f acc0 = {}, acc1 = {}, acc2 = {}, acc3 = {};

  for (int k0 = 0; k0 < DIM; k0 += 4) {
    v2f a = *(const v2f*)(xrow + k0 + hi * 2);               // 8B aligned
    const v2f* __restrict__ wp = Wp + (size_t)(k0 / 2 + hi) * DIM;
    v2f b0 = wp[ncol0     ];
    v2f b1 = wp[ncol0 + 16];
    v2f b2 = wp[ncol0 + 32];
    v2f b3 = wp[ncol0 + 48];
    acc0 = __builtin_amdgcn_wmma_f32_16x16x4_f32(false, a, false, b0, (short)0, acc0, false, false);
    acc1 = __builtin_amdgcn_wmma_f32_16x16x4_f32(false, a, false, b1, (short)0, acc1, false, false);
    acc2 = __builtin_amdgcn_wmma_f32_16x16x4_f32(false, a, false, b2, (short)0, acc2, false, false);
    acc3 = __builtin_amdgcn_wmma_f32_16x16x4_f32(false, a, false, b3, (short)0, acc3, false, false);
  }

  const int rbase = mtile * 16 + hi * 8;
  v8f accs[4] = { acc0, acc1, acc2, acc3 };
  #pragma unroll
  for (int j = 0; j < 4; ++j) {
    const int   c  = ncol0 + j * 16;
    const float bb = bias[c];
    #pragma unroll
    for (int r = 0; r < 8; ++r) {
      z[(size_t)(rbase + r) * DIM + c] = accs[j][r] + bb;
    }
  }
}

// ---------------------------------------------------------------------------
// Zero a float4 buffer
// ---------------------------------------------------------------------------
__global__ __launch_bounds__(256) void zero_f4(float4* __restrict__ p, int n4)
{
  int i = blockIdx.x * blockDim.x + threadIdx.x;
  if (i < n4) p[i] = make_float4(0.f, 0.f, 0.f, 0.f);
}

// ---------------------------------------------------------------------------
// Edge scatter: agg[dst[e]][:] += z[src[e]][:]   (one wave per edge)
// Gather via 2x global_load_b128 per lane; scatter via global_atomic_add_f32.
// ---------------------------------------------------------------------------
__global__ __launch_bounds__(256) void edge_scatter(
    const float* __restrict__ z, const int* __restrict__ src,
    const int* __restrict__ dst, float* __restrict__ agg, int n_edges)
{
  const int e    = (blockIdx.x * blockDim.x + threadIdx.x) >> 5;
  const int lane = threadIdx.x & 31;
  if (e >= n_edges) return;
  const int s = src[e];
  const int d = dst[e];
  const float4* __restrict__ zs4 = (const float4*)(z + (size_t)s * DIM);
  float* __restrict__        ad  = agg + (size_t)d * DIM;
  #pragma unroll
  for (int i = 0; i < 2; ++i) {
    float4 v  = zs4[lane + i * 32];          // 512B coalesced per step
    int    b  = (lane + i * 32) * 4;
    atomicAdd(&ad[b + 0], v.x);
    atomicAdd(&ad[b + 1], v.y);
    atomicAdd(&ad[b + 2], v.z);
    atomicAdd(&ad[b + 3], v.w);
  }
}

// ---------------------------------------------------------------------------
// Elementwise tanh in place (float4 vectorized)
// ---------------------------------------------------------------------------
__global__ __launch_bounds__(256) void tanh_inplace(float4* __restrict__ a, int n4)
{
  int i = blockIdx.x * blockDim.x + threadIdx.x;
  if (i < n4) {
    float4 v = a[i];
    v.x = tanhf(v.x); v.y = tanhf(v.y); v.z = tanhf(v.z); v.w = tanhf(v.w);
    a[i] = v;
  }
}

// ---------------------------------------------------------------------------
// Segment pooling scatter: sums[seg[n]][:] += h2[n][:]; counts[seg[n]] += 1
// ---------------------------------------------------------------------------
__global__ __launch_bounds__(256) void pool_scatter(
    const float* __restrict__ h2, const int* __restrict__ seg,
    float* __restrict__ sums, float* __restrict__ counts, int n_nodes)
{
  const int n    = (blockIdx.x * blockDim.x + threadIdx.x) >> 5;
  const int lane = threadIdx.x & 31;
  if (n >= n_nodes) return;
  const int s = seg[n];
  const float4* __restrict__ hr4 = (const float4*)(h2 + (size_t)n * DIM);
  float* __restrict__        sr  = sums + (size_t)s * DIM;
  #pragma unroll
  for (int i = 0; i < 2; ++i) {
    float4 v = hr4[lane + i * 32];
    int    b = (lane + i * 32) * 4;
    atomicAdd(&sr[b + 0], v.x);
    atomicAdd(&sr[b + 1], v.y);
    atomicAdd(&sr[b + 2], v.z);
    atomicAdd(&sr[b + 3], v.w);
  }
  if (lane == 0) atomicAdd(&counts[s], 1.0f);
}

// ---------------------------------------------------------------------------
// Finalize: out = sums / max(counts, 1)
// ---------------------------------------------------------------------------
__global__ __launch_bounds__(256) void pool_finalize(
    const float* __restrict__ sums, const float* __restrict__ counts,
    float* __restrict__ out)
{
  int i = blockIdx.x * blockDim.x + threadIdx.x;   // 800*256 elements
  if (i < NUM_SLOTS * DIM) {
    float c = counts[i >> 8];
    out[i] = sums[i] / fmaxf(c, 1.0f);
  }
}

// ---------------------------------------------------------------------------
extern "C" void kernel_launch(void* const* d_in, const int* in_sizes, int n_in,
                              void* d_out, int out_size, void* d_ws, size_t ws_size,
                              hipStream_t stream)
{
  const float* h    = (const float*)d_in[0];
  const float* W1   = (const float*)d_in[1];
  const float* b1   = (const float*)d_in[2];
  const float* W2   = (const float*)d_in[3];
  const float* b2   = (const float*)d_in[4];
  const int*   esrc = (const int*)d_in[5];
  const int*   edst = (const int*)d_in[6];
  const int*   seg  = (const int*)d_in[7];
  float*       out  = (float*)d_out;

  char* ws = (char*)d_ws;
  const size_t node_bytes = (size_t)N_NODES * DIM * sizeof(float);   // 102.4 MB
  float* bufA   = (float*)ws;                          // z1 / z2
  float* bufB   = (float*)(ws + node_bytes);           // agg1/h1 then agg2/h2
  float* sums   = (float*)(ws + 2 * node_bytes);       // 800*256
  float* counts = sums + NUM_SLOTS * DIM;              // 800
  size_t wp_off = 2 * node_bytes + (size_t)(NUM_SLOTS * DIM + NUM_SLOTS) * sizeof(float);
  wp_off        = (wp_off + 255) & ~(size_t)255;       // align 256B
  v2f* Wp1      = (v2f*)(ws + wp_off);                 // 256 KB
  v2f* Wp2      = Wp1 + (DIM / 2) * DIM;               // 256 KB

  const dim3 blk(256);
  const int gemm_blocks  = (N_NODES / 16) * 4 / 8;             // 3125
  const int node_f4      = N_NODES * DIM / 4;                   // 6.4M
  const int node_f4_blks = (node_f4 + 255) / 256;               // 25000
  const int edge_blocks  = (N_EDGES * 32 + 255) / 256;          // 200000
  const int pool_blocks  = (N_NODES * 32 + 255) / 256;          // 12500
  const int psum_f4      = (NUM_SLOTS * DIM + NUM_SLOTS) / 4;   // 51400
  const int psum_blks    = (psum_f4 + 255) / 256;
  const int pack_blocks  = ((DIM / 2) * DIM + 255) / 256;       // 128

  // ---- Weight repack (tiny) ----
  pack_W<<<pack_blocks, blk, 0, stream>>>(W1, Wp1);
  pack_W<<<pack_blocks, blk, 0, stream>>>(W2, Wp2);

  // ---- Layer 1 ----
  gemm_bias_wmma<<<gemm_blocks, blk, 0, stream>>>(h, Wp1, b1, bufA, N_NODES);
  zero_f4<<<node_f4_blks, blk, 0, stream>>>((float4*)bufB, node_f4);
  edge_scatter<<<edge_blocks, blk, 0, stream>>>(bufA, esrc, edst, bufB, N_EDGES);
  tanh_inplace<<<node_f4_blks, blk, 0, stream>>>((float4*)bufB, node_f4);   // bufB = h1

  // ---- Layer 2 ----
  gemm_bias_wmma<<<gemm_blocks, blk, 0, stream>>>(bufB, Wp2, b2, bufA, N_NODES); // bufA = z2
  zero_f4<<<node_f4_blks, blk, 0, stream>>>((float4*)bufB, node_f4);
  edge_scatter<<<edge_blocks, blk, 0, stream>>>(bufA, esrc, edst, bufB, N_EDGES);
  tanh_inplace<<<node_f4_blks, blk, 0, stream>>>((float4*)bufB, node_f4);   // bufB = h2

  // ---- Pooling ----
  zero_f4<<<psum_blks, blk, 0, stream>>>((float4*)sums, psum_f4);  // sums+counts contiguous
  pool_scatter<<<pool_blocks, blk, 0, stream>>>(bufB, seg, sums, counts, N_NODES);
  pool_finalize<<<(NUM_SLOTS * DIM + 255) / 256, blk, 0, stream>>>(sums, counts, out);
}